// LGCN_REL_EMB_89240830477004
// MI455X (gfx1250) — compile-verified
//
#include <hip/hip_runtime.h>
#include <hip/hip_bf16.h>

#define N_NODES 50000
#define RP 16
#define EMB 16
#define NCLS 40
#define NRP (N_NODES * RP)   // 800000
#define KTOT (RP * EMB)      // 256
#define KSTEPS (KTOT / 4)    // 64 wmma steps
#define NCT 3                // class tiles of 16 covering 40 (+8 pad)

typedef __attribute__((ext_vector_type(2))) float v2f;
typedef __attribute__((ext_vector_type(8))) float v8f;

__device__ __forceinline__ void atomAddF(float* a, float v) {
    // hardware global_atomic_add_f32
    unsafeAtomicAdd(a, v);
}

// ---------------- zero workspace ----------------
__global__ void k_zero(float* __restrict__ p, long count) {
    long i = (long)blockIdx.x * blockDim.x + threadIdx.x;
    long stride = (long)gridDim.x * blockDim.x;
    for (; i < count; i += stride) p[i] = 0.0f;
}

// ---------------- step 1: latents scatter ----------------
__global__ void k_rel_scatter(const int* __restrict__ rows, const int* __restrict__ cols,
                              const float* __restrict__ vals, const float* __restrict__ rel_emb,
                              float* __restrict__ latents, int nnz) {
    int j = blockIdx.x * blockDim.x + threadIdx.x;
    if (j >= nnz * RP) return;
    int i = j >> 4, p = j & 15;
    atomAddF(&latents[rows[i] * RP + p], vals[i] * rel_emb[cols[i] * RP + p]);
}

// ---------------- step 2: row softmax over 16 ----------------
__global__ void k_softmax(float* __restrict__ latents, int nt) {
    int t = blockIdx.x * blockDim.x + threadIdx.x;
    if (t >= nt) return;
    float* row = latents + (size_t)t * RP;
    float m = row[0];
#pragma unroll
    for (int k = 1; k < RP; ++k) m = fmaxf(m, row[k]);
    float e[RP], s = 0.0f;
#pragma unroll
    for (int k = 0; k < RP; ++k) { e[k] = __expf(row[k] - m); s += e[k]; }
    float inv = 1.0f / s;
#pragma unroll
    for (int k = 0; k < RP; ++k) row[k] = e[k] * inv;
}

// ---------------- step 3a: p==0 slice of colsum/rowsum (all hits bin 0) ----------------
__global__ __launch_bounds__(256) void k_p0_sums(const float* __restrict__ latents,
                                                 float* __restrict__ colsum,
                                                 float* __restrict__ rowsum, int nt) {
    __shared__ float s[256];
    float acc = 0.0f;
    for (int t = blockIdx.x * blockDim.x + threadIdx.x; t < nt; t += gridDim.x * blockDim.x)
        acc += latents[(size_t)t * RP];           // latents[t, 0]
    s[threadIdx.x] = acc;
    __syncthreads();
    for (int off = 128; off > 0; off >>= 1) {
        if (threadIdx.x < (unsigned)off) s[threadIdx.x] += s[threadIdx.x + off];
        __syncthreads();
    }
    if (threadIdx.x == 0) { atomAddF(&colsum[0], s[0]); atomAddF(&rowsum[0], s[0]); }
}

// ---------------- step 3b: p>=1 colsum/rowsum scatter (p = blockIdx.y+1) ----------------
__global__ void k_sums(const float* __restrict__ latents, const int* __restrict__ h_cols,
                       const int* __restrict__ v_rows, float* __restrict__ colsum,
                       float* __restrict__ rowsum, int nt) {
    int t = blockIdx.x * blockDim.x + threadIdx.x;
    if (t >= nt) return;
    int p = blockIdx.y + 1;
    int i = p * nt + t;
    float v = latents[(size_t)t * RP + p];
    atomAddF(&colsum[h_cols[i]], v);
    atomAddF(&rowsum[v_rows[i]], v);
}

// ---------------- step 4: h = segsum(ln * w1f[h_cols]) over h_rows (p = blockIdx.y) ----------------
__global__ void k_h_accum(const float* __restrict__ latents, const float* __restrict__ colsum,
                          const float* __restrict__ w1f, const int* __restrict__ h_rows,
                          const int* __restrict__ h_cols, float* __restrict__ h, int nt) {
    int j = blockIdx.x * blockDim.x + threadIdx.x;
    int t = j >> 4, e = j & 15;
    if (t >= nt) return;
    int p = blockIdx.y;
    int i = p * nt + t;
    int wrow = h_cols[i];
    float ln = latents[(size_t)t * RP + p] / colsum[wrow];
    atomAddF(&h[h_rows[i] * EMB + e], ln * w1f[(size_t)wrow * EMB + e]);
}

// ---------------- step 5: h = relu(h + bias1) ----------------
__global__ void k_bias_relu(float* __restrict__ h, const float* __restrict__ bias1) {
    int j = blockIdx.x * blockDim.x + threadIdx.x;
    if (j >= NRP) return;
    h[j] = fmaxf(h[j] + bias1[j & 15], 0.0f);
}

// ---------------- step 6a: p==0 slice of h2 (all hits row 0): block LDS reduce ----------------
__global__ __launch_bounds__(256) void k_h2_p0(const float* __restrict__ latents,
                                               const float* __restrict__ rowsum,
                                               const float* __restrict__ h,
                                               const int* __restrict__ v_cols,
                                               float* __restrict__ h2, int nt) {
    __shared__ float sacc[256 * EMB];
    float acc[EMB];
#pragma unroll
    for (int e = 0; e < EMB; ++e) acc[e] = 0.0f;
    float rs0 = rowsum[0];
    for (int t = blockIdx.x * blockDim.x + threadIdx.x; t < nt; t += gridDim.x * blockDim.x) {
        float w = latents[(size_t)t * RP] / rs0;          // ln2 for (p=0, t)
        const float* hr = h + (size_t)v_cols[t] * EMB;    // v_cols[t] = o_u[t]
#pragma unroll
        for (int e = 0; e < EMB; ++e) acc[e] += w * hr[e];
    }
#pragma unroll
    for (int e = 0; e < EMB; ++e) sacc[threadIdx.x * EMB + e] = acc[e];
    __syncthreads();
    for (int off = 128; off > 0; off >>= 1) {
        if (threadIdx.x < (unsigned)off) {
#pragma unroll
            for (int e = 0; e < EMB; ++e)
                sacc[threadIdx.x * EMB + e] += sacc[(threadIdx.x + off) * EMB + e];
        }
        __syncthreads();
    }
    if (threadIdx.x < EMB) atomAddF(&h2[threadIdx.x], sacc[threadIdx.x]);
}

// ---------------- step 6b: p>=1 h2 scatter (p = blockIdx.y+1) ----------------
__global__ void k_h2_accum(const float* __restrict__ latents, const float* __restrict__ rowsum,
                           const float* __restrict__ h, const int* __restrict__ v_rows,
                           const int* __restrict__ v_cols, float* __restrict__ h2, int nt) {
    int j = blockIdx.x * blockDim.x + threadIdx.x;
    int t = j >> 4, e = j & 15;
    if (t >= nt) return;
    int p = blockIdx.y + 1;
    int i = p * nt + t;
    int dst = v_rows[i];
    float ln2 = latents[(size_t)t * RP + p] / rowsum[dst];
    atomAddF(&h2[(size_t)dst * EMB + e], ln2 * h[(size_t)v_cols[i] * EMB + e]);
}

// ---------------- step 7 prep: fragment-ordered, zero-padded B ----------------
// bfrag[ct][step][lane] = { B[4*step + 2*half + 0, c], B[4*step + 2*half + 1, c] }
//   half = lane>>4, c = ct*16 + (lane&15);  c >= 40 -> 0 (pad columns)
// where B[k, c] = w2 flat (256, 40).  Total 3*64*32 v2f = 48 KB, L2-resident.
__global__ void k_bfrag(const float* __restrict__ w2, v2f* __restrict__ bfrag) {
    int j = blockIdx.x * blockDim.x + threadIdx.x;   // NCT*KSTEPS*32 = 6144
    if (j >= NCT * KSTEPS * 32) return;
    int lane = j & 31;
    int step = (j >> 5) & (KSTEPS - 1);
    int ct = j >> 11;
    int k = step * 4 + 2 * (lane >> 4);
    int c = ct * 16 + (lane & 15);
    v2f b = {0.0f, 0.0f};
    if (c < NCLS) {
        b.x = w2[(size_t)k * NCLS + c];
        b.y = w2[(size_t)(k + 1) * NCLS + c];
    }
    bfrag[j] = b;
}

// ---------------- step 7: out[n,c] = sum_{r,hh} w2[r,hh,c] * h2[r*N+n, hh] + bias2[c] ----------------
// GEMM 50000 x 256 x 40 via V_WMMA_F32_16X16X4_F32. One wave per 16-node tile; the wave
// keeps 3 independent accumulators (one per class tile) so each A fragment feeds 3
// back-to-back independent WMMAs (no D->C stall chain, A traffic amortized 3x).
// B fragments come branch-free from the pre-swizzled bfrag buffer.
__global__ __launch_bounds__(32) void k_out_wmma(const float* __restrict__ h2,
                                                 const v2f* __restrict__ bfrag,
                                                 const float* __restrict__ bias2,
                                                 float* __restrict__ out) {
    const int lane = threadIdx.x;
    const int tileM = blockIdx.x;
    const int half = lane >> 4;
    const int m = tileM * 16 + (lane & 15);

    v8f acc0 = {}, acc1 = {}, acc2 = {};
    const v2f* bf = bfrag + lane;
#pragma unroll 1
    for (int r = 0; r < RP; ++r) {
        const float* abase = h2 + ((size_t)r * N_NODES + m) * EMB + 2 * half;
        const v2f* bbase = bf + r * 4 * 32;
#pragma unroll
        for (int s4 = 0; s4 < 4; ++s4) {
            v2f a = *(const v2f*)(abase + 4 * s4);               // A[m, k..k+1]
            v2f b0 = bbase[(0 * KSTEPS + s4) * 32];
            v2f b1 = bbase[(1 * KSTEPS + s4) * 32];
            v2f b2 = bbase[(2 * KSTEPS + s4) * 32];
            acc0 = __builtin_amdgcn_wmma_f32_16x16x4_f32(false, a, false, b0,
                                                         (short)0, acc0, false, false);
            acc1 = __builtin_amdgcn_wmma_f32_16x16x4_f32(false, a, false, b1,
                                                         (short)0, acc1, false, false);
            acc2 = __builtin_amdgcn_wmma_f32_16x16x4_f32(false, a, false, b2,
                                                         (short)0, acc2, false, false);
        }
    }
    // D layout: VGPR v -> row (v + 8*half), col = lane&15
    const int mBase = tileM * 16 + half * 8;
#pragma unroll
    for (int ct = 0; ct < NCT; ++ct) {
        int c = ct * 16 + (lane & 15);
        if (c < NCLS) {
            float bv = bias2[c];
            const v8f acc = (ct == 0) ? acc0 : (ct == 1) ? acc1 : acc2;
#pragma unroll
            for (int v = 0; v < 8; ++v)
                out[(size_t)(mBase + v) * NCLS + c] = acc[v] + bv;
        }
    }
}

extern "C" void kernel_launch(void* const* d_in, const int* in_sizes, int n_in,
                              void* d_out, int out_size, void* d_ws, size_t ws_size,
                              hipStream_t stream) {
    const float* rel_emb = (const float*)d_in[0];
    const float* w1      = (const float*)d_in[1];   // (16,50000,16) == w1f (800000,16)
    const float* w2      = (const float*)d_in[2];   // (16,16,40)   == (256,40)
    const float* bias1   = (const float*)d_in[3];
    const float* bias2   = (const float*)d_in[4];
    const int*   rm_rows = (const int*)d_in[5];
    const int*   rm_cols = (const int*)d_in[6];
    const float* rm_vals = (const float*)d_in[7];
    const int*   h_rows  = (const int*)d_in[8];
    const int*   h_cols  = (const int*)d_in[9];
    const int*   v_rows  = (const int*)d_in[10];
    const int*   v_cols  = (const int*)d_in[11];
    float* out = (float*)d_out;

    const int nnz   = in_sizes[5];
    const int rp_nt = in_sizes[8];
    const int nt    = rp_nt / RP;

    float* ws      = (float*)d_ws;
    float* latents = ws;                               // nt*16
    float* colsum  = latents + (size_t)nt * RP;        // 800000
    float* rowsum  = colsum + NRP;                     // 800000
    float* hbuf    = rowsum + NRP;                     // 800000
    float* h2      = hbuf + NRP;                       // 12800000
    v2f*   bfrag   = (v2f*)(h2 + (size_t)RP * N_NODES * EMB);  // 6144 v2f
    const long zero_count = (long)nt * RP + 3L * NRP + (long)RP * N_NODES * EMB;

    k_zero<<<4096, 256, 0, stream>>>(ws, zero_count);

    {
        int thr = nnz * RP;
        k_rel_scatter<<<(thr + 255) / 256, 256, 0, stream>>>(rm_rows, rm_cols, rm_vals,
                                                             rel_emb, latents, nnz);
    }
    k_softmax<<<(nt + 255) / 256, 256, 0, stream>>>(latents, nt);

    k_p0_sums<<<512, 256, 0, stream>>>(latents, colsum, rowsum, nt);
    {
        dim3 g((nt + 255) / 256, RP - 1);
        k_sums<<<g, 256, 0, stream>>>(latents, h_cols, v_rows, colsum, rowsum, nt);
    }
    {
        dim3 g((nt * 16 + 255) / 256, RP);
        k_h_accum<<<g, 256, 0, stream>>>(latents, colsum, w1, h_rows, h_cols, hbuf, nt);
    }
    k_bias_relu<<<(NRP + 255) / 256, 256, 0, stream>>>(hbuf, bias1);

    k_h2_p0<<<256, 256, 0, stream>>>(latents, rowsum, hbuf, v_cols, h2, nt);
    {
        dim3 g((nt * 16 + 255) / 256, RP - 1);
        k_h2_accum<<<g, 256, 0, stream>>>(latents, rowsum, hbuf, v_rows, v_cols, h2, nt);
    }

    k_bfrag<<<(NCT * KSTEPS * 32 + 255) / 256, 256, 0, stream>>>(w2, bfrag);
    k_out_wmma<<<N_NODES / 16, 32, 0, stream>>>(h2, bfrag, bias2, out);
}